// GridQuantizer_9895604650083
// MI455X (gfx1250) — compile-verified
//
#include <hip/hip_runtime.h>

typedef __attribute__((ext_vector_type(2))) float v2f;
typedef __attribute__((ext_vector_type(8))) float v8f;

#define N_Q    8192
#define M_P    16384
#define WAVES  4
#define SLICE  (M_P / WAVES)   // 4096 protos per wave
#define UNROLL 4               // 4 WMMAs in flight per loop trip

// D = A x B + C with A rows = [-2x0, -2x1, 1, 0] (16x4) and
// B cols = [p0; p1; p0^2+p1^2; 0] (4x16):
//   s[m,n] = -2 x_m . p_n + |p_n|^2 = d2[m,n] - |x_m|^2.
// min/argmin over n of s matches min/argmin of dist (monotone transforms).
__global__ __launch_bounds__(128) void GridQuantizer_wmma_kernel(
    const float* __restrict__ x,       // [N_Q, 2]
    const float* __restrict__ protos,  // [M_P, 2]
    float* __restrict__ out_dist,      // [N_Q]
    int*   __restrict__ out_idx) {     // [N_Q]
  __shared__ float s_val[WAVES * 16];
  __shared__ int   s_idx[WAVES * 16];

  const int lane    = threadIdx.x & 31;
  const int wave    = threadIdx.x >> 5;
  const int half    = lane >> 4;      // wave32: lanes 0-15 vs 16-31
  const int l16     = lane & 15;
  const int rowBase = blockIdx.x * 16;

  // ---- A operand (constant over the whole scan) ----
  // 32-bit A 16x4 layout: lanes 0-15 hold K={0,1} for M=lane; lanes 16-31 K={2,3}.
  v2f a;
  {
    const int q    = rowBase + l16;
    const float x0 = x[2 * q + 0];
    const float x1 = x[2 * q + 1];
    a.x = (half == 0) ? (-2.0f * x0) : 1.0f;   // K=0 / K=2
    a.y = (half == 0) ? (-2.0f * x1) : 0.0f;   // K=1 / K=3
  }

  float minv[8];
  int   mini[8];
#pragma unroll
  for (int i = 0; i < 8; ++i) { minv[i] = 3.0e38f; mini[i] = 0; }

  // per-lane proto stream: column n = (chunk base) + l16, 8B per chunk
  const v2f* __restrict__ pp = (const v2f*)protos + wave * SLICE + l16;

  for (int j = 0; j < SLICE / 16; j += UNROLL) {
    // Unconditional speculative prefetch of the next UNROLL chunks
    // (silently dropped if it runs past the allocation on the last trip).
    __builtin_prefetch(pp + (j + UNROLL) * 16, 0, 1);

#pragma unroll
    for (int u = 0; u < UNROLL; ++u) {
      const int p = wave * SLICE + (j + u) * 16 + l16;  // candidate proto index
      const v2f pv = pp[(j + u) * 16];
      const float p0 = pv.x;
      const float p1 = pv.y;

      // Assumed 32-bit B 4x16 layout: VGPR r, lanes 0-15 -> K=r, lanes 16-31 -> K=r+2.
      v2f b;
      b.x = (half == 0) ? p0 : (p0 * p0 + p1 * p1);  // K=0 / K=2
      b.y = (half == 0) ? p1 : 0.0f;                 // K=1 / K=3

      v8f acc = {};
      acc = __builtin_amdgcn_wmma_f32_16x16x4_f32(
          /*neg_a=*/false, a, /*neg_b=*/false, b,
          /*c_mod=*/(short)0, acc, /*reuse_a=*/false, /*reuse_b=*/false);

      // C/D layout: VGPR i holds query m = i (+8 for lanes 16-31), column n = l16.
      // Strict < keeps the first (lowest-index) minimum along this lane's
      // monotonically increasing candidate stream.
#pragma unroll
      for (int i = 0; i < 8; ++i) {
        const bool t = acc[i] < minv[i];
        minv[i] = t ? acc[i] : minv[i];
        mini[i] = t ? p      : mini[i];
      }
    }
  }

  // ---- cross-lane (min, argmin) butterfly within each 16-lane half ----
  // Branch-free tie-break: smaller index wins on equal value.
#pragma unroll
  for (int m = 1; m <= 8; m <<= 1) {
#pragma unroll
    for (int i = 0; i < 8; ++i) {
      const float ov = __shfl_xor(minv[i], m, 32);
      const int   oi = __shfl_xor(mini[i], m, 32);
      const bool  t  = (ov < minv[i]) | ((ov == minv[i]) & (oi < mini[i]));
      minv[i] = t ? ov : minv[i];
      mini[i] = t ? oi : mini[i];
    }
  }

  // lane 0 holds queries rowBase+0..7, lane 16 holds rowBase+8..15
  if (l16 == 0) {
#pragma unroll
    for (int i = 0; i < 8; ++i) {
      s_val[wave * 16 + half * 8 + i] = minv[i];
      s_idx[wave * 16 + half * 8 + i] = mini[i];
    }
  }
  __syncthreads();

  // ---- combine the 4 wave slices (slices ordered => ties keep earlier idx) ----
  if (wave == 0 && lane < 16) {
    float best = s_val[lane];
    int   bidx = s_idx[lane];
#pragma unroll
    for (int w = 1; w < WAVES; ++w) {
      const float v  = s_val[w * 16 + lane];
      const int   ix = s_idx[w * 16 + lane];
      const bool  t  = (v < best) | ((v == best) & (ix < bidx));
      best = t ? v  : best;
      bidx = t ? ix : bidx;
    }
    const int q    = rowBase + lane;
    const float x0 = x[2 * q + 0];
    const float x1 = x[2 * q + 1];
    const float d2 = best + x0 * x0 + x1 * x1;
    out_dist[q] = sqrtf(fmaxf(d2, 0.0f));
    out_idx[q]  = bidx;
  }
}

extern "C" void kernel_launch(void* const* d_in, const int* in_sizes, int n_in,
                              void* d_out, int out_size, void* d_ws, size_t ws_size,
                              hipStream_t stream) {
  (void)in_sizes; (void)n_in; (void)out_size; (void)d_ws; (void)ws_size;
  const float* x      = (const float*)d_in[0];   // [8192, 2] f32
  const float* protos = (const float*)d_in[1];   // [16384, 2] f32
  float* out_dist = (float*)d_out;               // first N_Q floats: mindist
  int*   out_idx  = (int*)(out_dist + N_Q);      // next N_Q: argmin (int32 bits)

  dim3 grid(N_Q / 16);   // 512 row-tiles of 16 queries
  dim3 block(128);       // 4 waves x wave32, each scans 4096 protos
  hipLaunchKernelGGL(GridQuantizer_wmma_kernel, grid, block, 0, stream,
                     x, protos, out_dist, out_idx);
}